// RNSF_ContrastiveLoss_61649960566945
// MI455X (gfx1250) — compile-verified
//
#include <hip/hip_runtime.h>
#include <hip/hip_bf16.h>
#include <math.h>

// ---------------------------------------------------------------------------
// RNSF contrastive loss, MI455X (gfx1250, wave32).
// Key insight: top-k over the 8192x8192 mask collapses to 4 class-wise top-50
// selections. GEMM is [8192x128]x[128x256] (4 classes x 64 padded cols), done
// with full-precision V_WMMA_F32_16X16X4_F32 (problem is bandwidth-bound;
// precision matters more than matrix throughput here).
// ---------------------------------------------------------------------------

typedef float v2f __attribute__((ext_vector_type(2)));
typedef float v8f __attribute__((ext_vector_type(8)));

#define NPIX   8192          // 2*64*64
#define CCH    128
#define HW     4096          // 64*64
#define KSEL   50
#define CPAD   64            // padded columns per class
#define NCOL   256           // 4 classes * 64
#define INV_TAU (1.0f/0.07f)
#define LEPS   1e-8f

// ---------------- workspace layout (bytes, all 8B-aligned) ----------------
#define OFF_NIN      0u
#define OFF_NPOS     4194304u        // 8192*128*4
#define OFF_NNEG     8388608u
#define OFF_PSIM     12582912u
#define OFF_SEGIN    12615680u
#define OFF_SEGNEG   12648448u
#define OFF_NPROB    12681216u
#define OFF_TOPK     12713984u       // 4*64 ints
#define OFF_NEGT     12715008u       // 256*128 floats
#define OFF_BLKSUM   12846080u       // 512 floats

// ---- 1) L2-normalize feature maps: [2,128,64,64] -> row-major [8192,128] ----
__global__ __launch_bounds__(256) void k_normalize(
    const float* __restrict__ in, const float* __restrict__ pos,
    const float* __restrict__ neg,
    float* __restrict__ nin, float* __restrict__ npos, float* __restrict__ nneg)
{
    int t = blockIdx.y;
    const float* src = (t == 0) ? in : (t == 1) ? pos : neg;
    float* dst       = (t == 0) ? nin : (t == 1) ? npos : nneg;
    int n  = blockIdx.x * 256 + threadIdx.x;           // pixel id
    int b  = n >> 12;
    int hw = n & (HW - 1);
    const float* base = src + (size_t)b * CCH * HW + hw;
    float ss = 0.f;
#pragma unroll 8
    for (int c = 0; c < CCH; ++c) { float v = base[(size_t)c * HW]; ss += v * v; }
    float scale = 1.0f / fmaxf(sqrtf(ss), 1e-12f);
    float* drow = dst + (size_t)n * CCH;
#pragma unroll 8
    for (int c = 0; c < CCH; ++c) drow[c] = base[(size_t)c * HW] * scale;
}

// ---- 2) logits -> argmax class + max softmax prob ----
__global__ __launch_bounds__(256) void k_logits(
    const float* __restrict__ lin, const float* __restrict__ lneg,
    int* __restrict__ seg_in, int* __restrict__ seg_neg,
    float* __restrict__ nprob)
{
    int n  = blockIdx.x * 256 + threadIdx.x;
    int b  = n >> 12;
    int hw = n & (HW - 1);
    {
        const float* p = lin + (size_t)b * 4 * HW + hw;
        float m = p[0]; int am = 0;
#pragma unroll
        for (int c = 1; c < 4; ++c) { float v = p[(size_t)c * HW]; if (v > m) { m = v; am = c; } }
        seg_in[n] = am;
    }
    {
        const float* p = lneg + (size_t)b * 4 * HW + hw;
        float v0 = p[0], v1 = p[HW], v2 = p[2 * HW], v3 = p[3 * HW];
        float m = v0; int am = 0;
        if (v1 > m) { m = v1; am = 1; }
        if (v2 > m) { m = v2; am = 2; }
        if (v3 > m) { m = v3; am = 3; }
        float s = expf(v0 - m) + expf(v1 - m) + expf(v2 - m) + expf(v3 - m);
        seg_neg[n] = am;
        nprob[n]   = 1.0f / s;                 // exp(max-max)/sum = max softmax prob
    }
}

// ---- 3) pos_sim[n] = <nin[n], npos[n]> ; one wave per pixel ----
__global__ __launch_bounds__(256) void k_possim(
    const float* __restrict__ nin, const float* __restrict__ npos,
    float* __restrict__ psim)
{
    int wave = threadIdx.x >> 5, lane = threadIdx.x & 31;
    int n = blockIdx.x * 8 + wave;
    const float* a = nin + (size_t)n * CCH;
    const float* b = npos + (size_t)n * CCH;
    float d = 0.f;
#pragma unroll
    for (int c = lane; c < CCH; c += 32) d += a[c] * b[c];
#pragma unroll
    for (int m = 16; m; m >>= 1) d += __shfl_xor(d, m, 32);
    if (lane == 0) psim[n] = d;
}

// ---- 4) per-class top-50 negatives by prob (value desc, index asc) ----
__global__ __launch_bounds__(256) void k_topk(
    const int* __restrict__ seg_neg, const float* __restrict__ nprob,
    int* __restrict__ topk)
{
    __shared__ unsigned kv[NPIX];      // 32 KB: float bits (all probs > 0 -> monotonic)
    __shared__ unsigned rv[256];
    __shared__ int      ri[256];
    int c = blockIdx.x, tid = threadIdx.x;
    for (int i = tid; i < NPIX; i += 256)
        kv[i] = (seg_neg[i] != c) ? __float_as_uint(nprob[i]) : 0u;
    __syncthreads();
    for (int it = 0; it < KSEL; ++it) {
        unsigned best = 0u; int bidx = -1;
        for (int i = tid; i < NPIX; i += 256) {         // ascending -> strict > keeps lowest idx
            unsigned v = kv[i];
            if (v > best) { best = v; bidx = i; }
        }
        rv[tid] = best; ri[tid] = bidx;
        __syncthreads();
        for (int off = 128; off; off >>= 1) {
            if (tid < off) {
                bool take = (rv[tid + off] > rv[tid]) ||
                            (rv[tid + off] == rv[tid] && ri[tid + off] >= 0 &&
                             (unsigned)ri[tid + off] < (unsigned)ri[tid]);
                if (take) { rv[tid] = rv[tid + off]; ri[tid] = ri[tid + off]; }
            }
            __syncthreads();
        }
        if (tid == 0) { topk[c * CPAD + it] = ri[0]; kv[ri[0]] = 0u; }
        __syncthreads();
    }
    if (tid >= KSEL && tid < CPAD) topk[c * CPAD + tid] = -1;
}

// ---- 5) pack selected negatives into B matrix negT[256][128] (cols 50..63 = 0) ----
__global__ __launch_bounds__(128) void k_pack(
    const float* __restrict__ nneg, const int* __restrict__ topk,
    float* __restrict__ negT)
{
    int col = blockIdx.x;              // 0..255
    int k   = col & (CPAD - 1);
    float v = 0.f;
    if (k < KSEL) {
        int row = topk[col];
        v = nneg[(size_t)row * CCH + threadIdx.x];
    }
    negT[(size_t)col * CCH + threadIdx.x] = v;
}

// ---- 6) fused fp32-WMMA GEMM + masked softmax-denominator + per-row loss ----
// 512 blocks (16 rows each) x 16 waves (16 cols each). K=128 -> 32 WMMA/sim.
__global__ __launch_bounds__(512) void k_gemm_loss(
    const float* __restrict__ nin, const float* __restrict__ npos,
    const float* __restrict__ negT, const int* __restrict__ seg_in,
    const float* __restrict__ psim, float* __restrict__ blocksum)
{
    __shared__ float partial[16][16];  // [wave][local row]
    __shared__ float termsum[16];
    __shared__ int   segr[16];
    __shared__ float psimr[16];

    int tid  = threadIdx.x;
    int wave = tid >> 5, lane = tid & 31;
    int half = lane >> 4, l15 = lane & 15;
    int j0 = blockIdx.x * 16;          // global row tile
    int n0 = wave * 16;                // global col tile (covers 0..255)

    if (tid < 16) {
        segr[tid]    = seg_in[j0 + tid];
        psimr[tid]   = psim[j0 + tid];
        termsum[tid] = 0.f;
    }
    __syncthreads();

    // fragment base pointers:
    //   A (16x4 f32): lane l<16 -> row j0+l, K = k0 + {0,1}; lanes 16..31 -> K = k0 + {2,3}
    //   B (4x16 f32): lane l<16 -> col n0+l, K = k0 + {0,1}; lanes 16..31 -> K = k0 + {2,3}
    const float* brow = negT + (size_t)(n0 + l15) * CCH + 2 * half;

    for (int s = 0; s < 2; ++s) {
        const float* A = (s == 0) ? nin : npos;
        const float* arow = A + (size_t)(j0 + l15) * CCH + 2 * half;

        v8f acc = {};
#pragma unroll
        for (int k0 = 0; k0 < CCH; k0 += 4) {
            v2f a = *(const v2f*)(arow + k0);
            v2f b = *(const v2f*)(brow + k0);
            acc = __builtin_amdgcn_wmma_f32_16x16x4_f32(
                false, a, false, b, (short)0, acc, false, false);
        }

        // C layout: VGPR v, lanes 0-15 -> row v, lanes 16-31 -> row v+8; col = l15.
        int col = n0 + l15;
#pragma unroll
        for (int v = 0; v < 8; ++v) {
            int r   = v + 8 * half;
            int lo  = segr[r] * CPAD;
            bool in = (col >= lo) & (col < lo + KSEL);
            float e = in ? __expf(acc[v] * INV_TAU) : 0.f;
#pragma unroll
            for (int m = 8; m; m >>= 1) e += __shfl_xor(e, m, 16);
            if (l15 == 0) partial[wave][r] = e;
        }
        __syncthreads();

        if (tid < 16) {
            float den = 0.f;
#pragma unroll
            for (int w = 0; w < 16; ++w) den += partial[w][tid];
            float nom = __expf(psimr[tid] * INV_TAU);
            termsum[tid] += logf(den + nom + LEPS) - psimr[tid] * INV_TAU;
        }
        __syncthreads();
    }

    if (tid == 0) {
        float s = 0.f;
#pragma unroll
        for (int r = 0; r < 16; ++r) s += termsum[r];
        blocksum[blockIdx.x] = s;
    }
}

// ---- 7) deterministic final reduction ----
__global__ __launch_bounds__(512) void k_final(
    const float* __restrict__ blocksum, float* __restrict__ out)
{
    __shared__ float s[512];
    int tid = threadIdx.x;
    s[tid] = blocksum[tid];
    __syncthreads();
    for (int off = 256; off; off >>= 1) {
        if (tid < off) s[tid] += s[tid + off];
        __syncthreads();
    }
    if (tid == 0) out[0] = s[0] * (1.0f / (float)NPIX);
}

extern "C" void kernel_launch(void* const* d_in, const int* in_sizes, int n_in,
                              void* d_out, int out_size, void* d_ws, size_t ws_size,
                              hipStream_t stream) {
    const float* in   = (const float*)d_in[0];
    const float* pos  = (const float*)d_in[1];
    const float* neg  = (const float*)d_in[2];
    const float* lin  = (const float*)d_in[3];
    const float* lneg = (const float*)d_in[4];
    float* out = (float*)d_out;

    char* ws = (char*)d_ws;
    float* nin     = (float*)(ws + OFF_NIN);
    float* npos    = (float*)(ws + OFF_NPOS);
    float* nneg    = (float*)(ws + OFF_NNEG);
    float* psim    = (float*)(ws + OFF_PSIM);
    int*   seg_in  = (int*)  (ws + OFF_SEGIN);
    int*   seg_neg = (int*)  (ws + OFF_SEGNEG);
    float* nprob   = (float*)(ws + OFF_NPROB);
    int*   topk    = (int*)  (ws + OFF_TOPK);
    float* negT    = (float*)(ws + OFF_NEGT);
    float* blksum  = (float*)(ws + OFF_BLKSUM);

    k_normalize<<<dim3(NPIX / 256, 3), 256, 0, stream>>>(in, pos, neg, nin, npos, nneg);
    k_logits   <<<NPIX / 256, 256, 0, stream>>>(lin, lneg, seg_in, seg_neg, nprob);
    k_possim   <<<NPIX / 8,   256, 0, stream>>>(nin, npos, psim);
    k_topk     <<<4,          256, 0, stream>>>(seg_neg, nprob, topk);
    k_pack     <<<NCOL,       128, 0, stream>>>(nneg, topk, negT);
    k_gemm_loss<<<NPIX / 16,  512, 0, stream>>>(nin, npos, negT, seg_in, psim, blksum);
    k_final    <<<1,          512, 0, stream>>>(blksum, out);
}